// HiGATE_85873576117155
// MI455X (gfx1250) — compile-verified
//
#include <hip/hip_runtime.h>
#include <math.h>

// ---------------- problem constants ----------------
#define NCELL   60000
#define FDIM    256
#define HDIM    128
#define NHEAD   8
#define DHEAD   16
#define NLAYER  2
#define CCLUST  256
#define EEDGE   600000
#define ETEDGE  1500
#define NCLSOUT 4

typedef __attribute__((ext_vector_type(2))) float v2f;
typedef __attribute__((ext_vector_type(8))) float v8f;

// ---------------- helpers ----------------
__device__ __forceinline__ void atomic_max_f(float* addr, float val) {
  // signed-float atomic max: int-max for non-negative, uint-min for negative.
  if (val >= 0.0f) atomicMax((int*)addr, __float_as_int(val));
  else             atomicMin((unsigned int*)addr, __float_as_uint(val));
}
__device__ __forceinline__ float sigmf(float x) { return 1.0f / (1.0f + __expf(-x)); }

// ---------------- fp32 WMMA GEMM: Y[M,N] = X[M,K] @ W[N,K]^T (+bias) (+relu) ------------------
// One wave32 computes a 16x64 output tile (4 accumulators) with V_WMMA_F32_16X16X4_F32.
// The A fragment (activations) is loaded once per K-step and reused by 4 WMMAs. The K-loop is
// explicitly software-pipelined (load next fragments, then issue current WMMAs) so the VMEM
// latency of step k+1 is hidden under the 4 matrix ops of step k.
__global__ void k_wmma_gemm(const float* __restrict__ X, const float* __restrict__ W,
                            const float* __restrict__ bias, float* __restrict__ Y,
                            int M, int K, int Nout, int relu) {
  int wave   = (blockIdx.x * blockDim.x + threadIdx.x) >> 5;
  int lane   = threadIdx.x & 31;
  int tilesN = Nout >> 6;                       // 64-wide N tiles
  int tiles  = (M >> 4) * tilesN;
  if (wave >= tiles) return;                    // wave-uniform: EXEC stays all-ones below
  int m0 = (wave / tilesN) << 4;
  int n0 = (wave % tilesN) << 6;

  int row   = lane & 15;
  int khalf = (lane >> 4) << 1;                 // 0 for lanes 0-15, 2 for lanes 16-31
  const float* xp  = X + (size_t)(m0 + row) * K + khalf;           // A: 16x4 fragment
  const float* wp0 = W + (size_t)(n0 + row) * K + khalf;           // B: 4x16 fragments
  const float* wp1 = wp0 + (size_t)16 * K;
  const float* wp2 = wp1 + (size_t)16 * K;
  const float* wp3 = wp2 + (size_t)16 * K;

  v8f acc0 = {}, acc1 = {}, acc2 = {}, acc3 = {};

  // prologue: fragments for k = 0
  v2f a  = *(const v2f*)(xp);
  v2f b0 = *(const v2f*)(wp0);
  v2f b1 = *(const v2f*)(wp1);
  v2f b2 = *(const v2f*)(wp2);
  v2f b3 = *(const v2f*)(wp3);

#pragma unroll 4
  for (int k = 4; k < K; k += 4) {
    // issue next iteration's loads first, then consume current fragments
    v2f an  = *(const v2f*)(xp + k);
    v2f b0n = *(const v2f*)(wp0 + k);
    v2f b1n = *(const v2f*)(wp1 + k);
    v2f b2n = *(const v2f*)(wp2 + k);
    v2f b3n = *(const v2f*)(wp3 + k);
    acc0 = __builtin_amdgcn_wmma_f32_16x16x4_f32(false, a, false, b0, (short)0, acc0, false, false);
    acc1 = __builtin_amdgcn_wmma_f32_16x16x4_f32(false, a, false, b1, (short)0, acc1, false, false);
    acc2 = __builtin_amdgcn_wmma_f32_16x16x4_f32(false, a, false, b2, (short)0, acc2, false, false);
    acc3 = __builtin_amdgcn_wmma_f32_16x16x4_f32(false, a, false, b3, (short)0, acc3, false, false);
    a = an; b0 = b0n; b1 = b1n; b2 = b2n; b3 = b3n;
  }
  // epilogue: last K-step
  acc0 = __builtin_amdgcn_wmma_f32_16x16x4_f32(false, a, false, b0, (short)0, acc0, false, false);
  acc1 = __builtin_amdgcn_wmma_f32_16x16x4_f32(false, a, false, b1, (short)0, acc1, false, false);
  acc2 = __builtin_amdgcn_wmma_f32_16x16x4_f32(false, a, false, b2, (short)0, acc2, false, false);
  acc3 = __builtin_amdgcn_wmma_f32_16x16x4_f32(false, a, false, b3, (short)0, acc3, false, false);

  // D layout: VGPR r -> lanes 0-15: (m0+r, n0+lane); lanes 16-31: (m0+r+8, n0+lane-16)
  int rbase = m0 + ((lane >> 4) << 3);
  v8f accs[4] = {acc0, acc1, acc2, acc3};
#pragma unroll
  for (int t = 0; t < 4; ++t) {
    int col = n0 + (t << 4) + (lane & 15);
    float bv = bias ? bias[col] : 0.0f;
#pragma unroll
    for (int r = 0; r < 8; ++r) {
      float v = accs[t][r] + bv;
      if (relu) v = fmaxf(v, 0.0f);
      Y[(size_t)(rbase + r) * Nout + col] = v;
    }
  }
}

static inline void gemm(const float* X, const float* W, const float* b, float* Y,
                        int M, int K, int Nout, int relu, hipStream_t s) {
  int tiles  = (M / 16) * (Nout / 64);
  int blocks = (tiles + 7) / 8;                 // 8 waves (256 threads) per block
  k_wmma_gemm<<<blocks, 256, 0, s>>>(X, W, b, Y, M, K, Nout, relu);
}

// ---------------- small utility kernels ----------------
__global__ void k_fill(float* p, int n, float v) {
  int i = blockIdx.x * blockDim.x + threadIdx.x;
  if (i < n) p[i] = v;
}
__global__ void k_cluster_cnt(const int* __restrict__ lab, float* cnt) {
  int i = blockIdx.x * blockDim.x + threadIdx.x;
  if (i < NCELL) atomicAdd(&cnt[lab[i]], 1.0f);
}
__global__ void k_cluster_accum(const float* __restrict__ cf, const int* __restrict__ lab,
                                float* tx) {
  int i = blockIdx.x * blockDim.x + threadIdx.x;
  if (i >= NCELL * FDIM) return;
  int n = i / FDIM, f = i - n * FDIM;
  atomicAdd(&tx[lab[n] * FDIM + f], cf[i]);
}
__global__ void k_cluster_mean(float* tx, const float* __restrict__ cnt) {
  int i = blockIdx.x * blockDim.x + threadIdx.x;
  if (i < CCLUST * FDIM) tx[i] /= fmaxf(cnt[i / FDIM], 1.0f);
}
__global__ void k_tissue_deg(const int* __restrict__ tdst, float* degc) {
  int i = blockIdx.x * blockDim.x + threadIdx.x;
  if (i >= ETEDGE + CCLUST) return;
  int d = (i < ETEDGE) ? tdst[i] : (i - ETEDGE);
  atomicAdd(&degc[d], 1.0f);
}
__global__ void k_dinv(const float* __restrict__ degc, float* dinv) {
  int i = blockIdx.x * blockDim.x + threadIdx.x;
  if (i < CCLUST) dinv[i] = degc[i] > 0.0f ? rsqrtf(degc[i]) : 0.0f;
}

// ---------------- GAT edge kernels ----------------
__global__ void k_gat_scores(const float* __restrict__ h, const float* __restrict__ asrc,
                             const float* __restrict__ adst, float* a_s, float* a_d) {
  int i = blockIdx.x * blockDim.x + threadIdx.x;
  if (i >= NCELL * NHEAD) return;
  int n = i / NHEAD, hd = i - n * NHEAD;
  const float* hp = h + (size_t)n * HDIM + hd * DHEAD;
  float s = 0.0f, d = 0.0f;
#pragma unroll
  for (int k = 0; k < DHEAD; ++k) {
    float v = hp[k];
    s += v * asrc[hd * DHEAD + k];
    d += v * adst[hd * DHEAD + k];
  }
  a_s[i] = s; a_d[i] = d;
}
__device__ __forceinline__ float edge_score(const int* esrc, const int* edst, int e, int hd,
                                            const float* a_s, const float* a_d, int* so, int* dob) {
  int s = (e < EEDGE) ? esrc[e] : (e - EEDGE);
  int d = (e < EEDGE) ? edst[e] : (e - EEDGE);
  *so = s; *dob = d;
  float x = a_s[s * NHEAD + hd] + a_d[d * NHEAD + hd];
  return x > 0.0f ? x : 0.2f * x;              // leaky_relu(0.2)
}
__global__ void k_edge_max(const int* __restrict__ esrc, const int* __restrict__ edst,
                           const float* __restrict__ a_s, const float* __restrict__ a_d,
                           float* m) {
  int i = blockIdx.x * blockDim.x + threadIdx.x;
  if (i >= (EEDGE + NCELL) * NHEAD) return;
  int e = i / NHEAD, hd = i - e * NHEAD, s, d;
  float x = edge_score(esrc, edst, e, hd, a_s, a_d, &s, &d);
  atomic_max_f(&m[d * NHEAD + hd], x);
}
__global__ void k_edge_expsum(const int* __restrict__ esrc, const int* __restrict__ edst,
                              const float* __restrict__ a_s, const float* __restrict__ a_d,
                              const float* __restrict__ m, float* z) {
  int i = blockIdx.x * blockDim.x + threadIdx.x;
  if (i >= (EEDGE + NCELL) * NHEAD) return;
  int e = i / NHEAD, hd = i - e * NHEAD, s, d;
  float x = edge_score(esrc, edst, e, hd, a_s, a_d, &s, &d);
  atomicAdd(&z[d * NHEAD + hd], __expf(x - m[d * NHEAD + hd]));
}
__global__ void k_edge_agg(const int* __restrict__ esrc, const int* __restrict__ edst,
                           const float* __restrict__ a_s, const float* __restrict__ a_d,
                           const float* __restrict__ m, const float* __restrict__ z,
                           const float* __restrict__ hbuf, float* agg) {
  int i = blockIdx.x * blockDim.x + threadIdx.x;
  if (i >= (EEDGE + NCELL) * NHEAD) return;
  int e = i / NHEAD, hd = i - e * NHEAD, s, d;
  float x = edge_score(esrc, edst, e, hd, a_s, a_d, &s, &d);
  float alpha = __expf(x - m[d * NHEAD + hd]) / z[d * NHEAD + hd];
  const float* hs = hbuf + (size_t)s * HDIM + hd * DHEAD;
  float* ag = agg + (size_t)d * HDIM + hd * DHEAD;
#pragma unroll
  for (int k = 0; k < DHEAD; ++k) atomicAdd(&ag[k], hs[k] * alpha);
}
__global__ void k_bias_elu(const float* __restrict__ agg, const float* __restrict__ b,
                           float* cell_h) {
  int i = blockIdx.x * blockDim.x + threadIdx.x;
  if (i >= NCELL * HDIM) return;
  float x = agg[i] + b[i % HDIM];
  cell_h[i] = x > 0.0f ? x : (__expf(x) - 1.0f);    // ELU
}

// ---------------- GCN (tissue graph) ----------------
__global__ void k_gcn_scatter(const int* __restrict__ tsrc, const int* __restrict__ tdst,
                              const float* __restrict__ dinv, const float* __restrict__ th,
                              float* out) {
  int i = blockIdx.x * blockDim.x + threadIdx.x;
  if (i >= (ETEDGE + CCLUST) * HDIM) return;
  int te = i / HDIM, c = i - te * HDIM;
  int s = (te < ETEDGE) ? tsrc[te] : (te - ETEDGE);
  int d = (te < ETEDGE) ? tdst[te] : (te - ETEDGE);
  atomicAdd(&out[d * HDIM + c], th[s * HDIM + c] * dinv[s] * dinv[d]);
}
__global__ void k_relu_bias(const float* __restrict__ x, const float* __restrict__ b,
                            float* y, int rows) {
  int i = blockIdx.x * blockDim.x + threadIdx.x;
  if (i < rows * HDIM) y[i] = fmaxf(x[i] + b[i % HDIM], 0.0f);
}

// ---------------- bottom-up cross-level attention ----------------
__global__ void k_bu_satt(const float* __restrict__ tq, const float* __restrict__ kbuf,
                          const int* __restrict__ lab, float* sbuf) {
  int i = blockIdx.x * blockDim.x + threadIdx.x;
  if (i >= NCELL * NHEAD) return;
  int n = i / NHEAD, hd = i - n * NHEAD;
  const float* qp = tq + (size_t)lab[n] * HDIM + hd * DHEAD;
  const float* kp = kbuf + (size_t)n * HDIM + hd * DHEAD;
  float s = 0.0f;
#pragma unroll
  for (int k = 0; k < DHEAD; ++k) s += qp[k] * kp[k];
  sbuf[i] = s * 0.25f;                          // scale = 1/sqrt(16)
}
__global__ void k_seg_max(const float* __restrict__ sbuf, const int* __restrict__ lab,
                          float* m2) {
  int i = blockIdx.x * blockDim.x + threadIdx.x;
  if (i >= NCELL * NHEAD) return;
  int n = i / NHEAD, hd = i - n * NHEAD;
  atomic_max_f(&m2[lab[n] * NHEAD + hd], sbuf[i]);
}
__global__ void k_seg_expsum(float* sbuf, const int* __restrict__ lab,
                             const float* __restrict__ m2, float* z2) {
  int i = blockIdx.x * blockDim.x + threadIdx.x;
  if (i >= NCELL * NHEAD) return;
  int n = i / NHEAD, hd = i - n * NHEAD;
  float w = __expf(sbuf[i] - m2[lab[n] * NHEAD + hd]);
  sbuf[i] = w;
  atomicAdd(&z2[lab[n] * NHEAD + hd], w);
}
__global__ void k_bu_ctx(const float* __restrict__ sbuf, const float* __restrict__ z2,
                         const int* __restrict__ lab, const float* __restrict__ vbuf,
                         float* tctx) {
  int i = blockIdx.x * blockDim.x + threadIdx.x;
  if (i >= NCELL * NHEAD) return;
  int n = i / NHEAD, hd = i - n * NHEAD;
  int c = lab[n];
  float alpha = sbuf[i] / z2[c * NHEAD + hd];
  const float* vp = vbuf + (size_t)n * HDIM + hd * DHEAD;
  float* tp = tctx + (size_t)c * HDIM + hd * DHEAD;
#pragma unroll
  for (int k = 0; k < DHEAD; ++k) atomicAdd(&tp[k], vp[k] * alpha);
}

// ---------------- cluster-level GRU (exact: GRU inputs are cluster-constant) ----------------
__global__ void k_xcat(const float* __restrict__ ttd, const float* __restrict__ th,
                       float* xcat) {
  int i = blockIdx.x * blockDim.x + threadIdx.x;
  if (i >= CCLUST * 2 * HDIM) return;
  int c = i / (2 * HDIM), j = i - c * 2 * HDIM;
  xcat[i] = (j < HDIM) ? ttd[c * HDIM + j] : th[c * HDIM + j - HDIM];
}
__global__ void k_gru(const float* __restrict__ gi, const float* __restrict__ gh,
                      const float* __restrict__ hprev, float* G) {
  int i = blockIdx.x * blockDim.x + threadIdx.x;
  if (i >= CCLUST * HDIM) return;
  int c = i / HDIM, j = i - c * HDIM;
  const float* gip = gi + c * 3 * HDIM;
  const float* ghp = gh + c * 3 * HDIM;
  float r  = sigmf(gip[j] + ghp[j]);
  float zz = sigmf(gip[HDIM + j] + ghp[HDIM + j]);
  float nn = tanhf(gip[2 * HDIM + j] + r * ghp[2 * HDIM + j]);
  G[i] = (1.0f - zz) * nn + zz * hprev[i];
}
__global__ void k_gather(const float* __restrict__ G, const int* __restrict__ lab,
                         float* cell_h) {
  int i = blockIdx.x * blockDim.x + threadIdx.x;
  if (i >= NCELL * HDIM) return;
  cell_h[i] = G[lab[i / HDIM] * HDIM + (i % HDIM)];
}

// ---------------- pooling + classifier head ----------------
__global__ void k_pool(const float* __restrict__ X, int rows, float inv_rows,
                       float* mean_out, float* max_out) {
  int c = blockIdx.x;                           // one block per channel (HDIM blocks)
  __shared__ float ssum[256], smax[256];
  float s = 0.0f, mx = -INFINITY;
  for (int r = threadIdx.x; r < rows; r += blockDim.x) {
    float v = X[(size_t)r * HDIM + c];
    s += v; mx = fmaxf(mx, v);
  }
  ssum[threadIdx.x] = s; smax[threadIdx.x] = mx;
  __syncthreads();
  for (int off = 128; off > 0; off >>= 1) {
    if (threadIdx.x < off) {
      ssum[threadIdx.x] += ssum[threadIdx.x + off];
      smax[threadIdx.x] = fmaxf(smax[threadIdx.x], smax[threadIdx.x + off]);
    }
    __syncthreads();
  }
  if (threadIdx.x == 0) { mean_out[c] = ssum[0] * inv_rows; max_out[c] = smax[0]; }
}
__global__ void k_mlp(const float* __restrict__ emb,
                      const float* __restrict__ W1, const float* __restrict__ b1,
                      const float* __restrict__ W2, const float* __restrict__ b2,
                      const float* __restrict__ W3, const float* __restrict__ b3,
                      float* out) {
  __shared__ float se[4 * HDIM], s1[2 * HDIM], s2[HDIM];
  int t = threadIdx.x;                          // 256 threads
  for (int j = t; j < 4 * HDIM; j += 256) se[j] = emb[j];
  __syncthreads();
  { float acc = b1[t]; const float* w = W1 + t * 4 * HDIM;
    for (int k = 0; k < 4 * HDIM; ++k) acc += w[k] * se[k];
    s1[t] = fmaxf(acc, 0.0f); }
  __syncthreads();
  if (t < HDIM) {
    float acc = b2[t]; const float* w = W2 + t * 2 * HDIM;
    for (int k = 0; k < 2 * HDIM; ++k) acc += w[k] * s1[k];
    s2[t] = fmaxf(acc, 0.0f);
  }
  __syncthreads();
  if (t < NCLSOUT) {
    float acc = b3[t]; const float* w = W3 + t * HDIM;
    for (int k = 0; k < HDIM; ++k) acc += w[k] * s2[k];
    out[t] = acc;
  }
}

// ---------------- host side ----------------
static inline float* carve(float** p, size_t n) {
  float* r = *p;
  *p += (n + 63) & ~(size_t)63;                 // 256B alignment
  return r;
}
static inline dim3 g1(long total, int bs = 256) { return dim3((unsigned)((total + bs - 1) / bs)); }

extern "C" void kernel_launch(void* const* d_in, const int* in_sizes, int n_in,
                              void* d_out, int out_size, void* d_ws, size_t ws_size,
                              hipStream_t stream) {
  const float* cell_features = (const float*)d_in[0];
  const int*   cell_edge     = (const int*)d_in[1];     // [2,E]
  const int*   lab           = (const int*)d_in[2];
  const int*   tissue_edge   = (const int*)d_in[3];     // [2,ET]
  const float* W_cell_proj   = (const float*)d_in[4];
  const float* b_cell_proj   = (const float*)d_in[5];
  const float* W_tissue_proj = (const float*)d_in[6];
  const float* b_tissue_proj = (const float*)d_in[7];
  const float* gat_W    = (const float*)d_in[8];
  const float* gat_asrc = (const float*)d_in[9];
  const float* gat_adst = (const float*)d_in[10];
  const float* gat_b    = (const float*)d_in[11];
  const float* gcn_W    = (const float*)d_in[12];
  const float* gcn_b    = (const float*)d_in[13];
  const float* bu_Wq = (const float*)d_in[14]; const float* bu_Wk = (const float*)d_in[15];
  const float* bu_Wv = (const float*)d_in[16];
  const float* bu_bq = (const float*)d_in[17]; const float* bu_bk = (const float*)d_in[18];
  const float* bu_bv = (const float*)d_in[19];
  const float* bu_Wo = (const float*)d_in[20]; const float* bu_bo = (const float*)d_in[21];
  const float* td_Wv = (const float*)d_in[22]; const float* td_bv = (const float*)d_in[23];
  const float* td_Wo = (const float*)d_in[24]; const float* td_bo = (const float*)d_in[25];
  const float* gru_Wih = (const float*)d_in[26]; const float* gru_bih = (const float*)d_in[27];
  const float* gru_Whh = (const float*)d_in[28]; const float* gru_bhh = (const float*)d_in[29];
  const float* Wc1 = (const float*)d_in[30]; const float* bc1 = (const float*)d_in[31];
  const float* Wc2 = (const float*)d_in[32]; const float* bc2 = (const float*)d_in[33];
  const float* Wc3 = (const float*)d_in[34]; const float* bc3 = (const float*)d_in[35];

  const int* esrc = cell_edge;            const int* edst = cell_edge + EEDGE;
  const int* tsrc = tissue_edge;          const int* tdst = tissue_edge + ETEDGE;

  // ---- workspace carve (~101 MB of floats) ----
  float* p = (float*)d_ws;
  float* cell_h = carve(&p, (size_t)NCELL * HDIM);
  float* bufA   = carve(&p, (size_t)NCELL * HDIM);
  float* bufB   = carve(&p, (size_t)NCELL * HDIM);
  float* a_s    = carve(&p, (size_t)NCELL * NHEAD);   // also reused as bottom-up sbuf
  float* a_d    = carve(&p, (size_t)NCELL * NHEAD);
  float* segm   = carve(&p, (size_t)NCELL * NHEAD);
  float* segz   = carve(&p, (size_t)NCELL * NHEAD);
  float* cnt    = carve(&p, CCLUST);
  float* tiss_x = carve(&p, CCLUST * FDIM);
  float* tiss_h = carve(&p, CCLUST * HDIM);
  float* tA     = carve(&p, CCLUST * HDIM);
  float* tB     = carve(&p, CCLUST * HDIM);
  float* tq     = carve(&p, CCLUST * HDIM);
  float* tctx   = carve(&p, CCLUST * HDIM);
  float* ttd    = carve(&p, CCLUST * HDIM);
  float* xcat   = carve(&p, CCLUST * 2 * HDIM);
  float* tgi    = carve(&p, CCLUST * 3 * HDIM);
  float* tgh    = carve(&p, CCLUST * 3 * HDIM);
  float* G      = carve(&p, CCLUST * HDIM);
  float* degc   = carve(&p, CCLUST);
  float* dinv   = carve(&p, CCLUST);
  float* m2     = carve(&p, CCLUST * NHEAD);
  float* z2     = carve(&p, CCLUST * NHEAD);
  float* emb    = carve(&p, 4 * HDIM);
  (void)ws_size; (void)in_sizes; (void)n_in; (void)out_size;

  const long eTotH = (long)(EEDGE + NCELL) * NHEAD;

  // ---- cluster means of raw features + tissue degrees ----
  hipMemsetAsync(cnt, 0, CCLUST * sizeof(float), stream);
  hipMemsetAsync(tiss_x, 0, CCLUST * FDIM * sizeof(float), stream);
  hipMemsetAsync(degc, 0, CCLUST * sizeof(float), stream);
  k_cluster_cnt<<<g1(NCELL), 256, 0, stream>>>(lab, cnt);
  k_cluster_accum<<<g1((long)NCELL * FDIM), 256, 0, stream>>>(cell_features, lab, tiss_x);
  k_cluster_mean<<<g1(CCLUST * FDIM), 256, 0, stream>>>(tiss_x, cnt);
  k_tissue_deg<<<g1(ETEDGE + CCLUST), 256, 0, stream>>>(tdst, degc);
  k_dinv<<<g1(CCLUST), 256, 0, stream>>>(degc, dinv);

  // ---- input projections (WMMA) ----
  gemm(cell_features, W_cell_proj, b_cell_proj, cell_h, NCELL, FDIM, HDIM, 0, stream);
  gemm(tiss_x, W_tissue_proj, b_tissue_proj, tiss_h, CCLUST, FDIM, HDIM, 0, stream);

  for (int l = 0; l < NLAYER; ++l) {
    // ---- GATConv + ELU ----
    gemm(cell_h, gat_W + (size_t)l * HDIM * HDIM, nullptr, bufA, NCELL, HDIM, HDIM, 0, stream);
    k_gat_scores<<<g1((long)NCELL * NHEAD), 256, 0, stream>>>(
        bufA, gat_asrc + l * NHEAD * DHEAD, gat_adst + l * NHEAD * DHEAD, a_s, a_d);
    k_fill<<<g1((long)NCELL * NHEAD), 256, 0, stream>>>(segm, NCELL * NHEAD, -INFINITY);
    hipMemsetAsync(segz, 0, (size_t)NCELL * NHEAD * sizeof(float), stream);
    hipMemsetAsync(bufB, 0, (size_t)NCELL * HDIM * sizeof(float), stream);
    k_edge_max<<<g1(eTotH), 256, 0, stream>>>(esrc, edst, a_s, a_d, segm);
    k_edge_expsum<<<g1(eTotH), 256, 0, stream>>>(esrc, edst, a_s, a_d, segm, segz);
    k_edge_agg<<<g1(eTotH), 256, 0, stream>>>(esrc, edst, a_s, a_d, segm, segz, bufA, bufB);
    k_bias_elu<<<g1((long)NCELL * HDIM), 256, 0, stream>>>(bufB, gat_b + l * HDIM, cell_h);

    // ---- GCNConv + ReLU on tissue graph ----
    gemm(tiss_h, gcn_W + (size_t)l * HDIM * HDIM, nullptr, tA, CCLUST, HDIM, HDIM, 0, stream);
    hipMemsetAsync(tB, 0, CCLUST * HDIM * sizeof(float), stream);
    k_gcn_scatter<<<g1((long)(ETEDGE + CCLUST) * HDIM), 256, 0, stream>>>(tsrc, tdst, dinv, tA, tB);
    k_relu_bias<<<g1(CCLUST * HDIM), 256, 0, stream>>>(tB, gcn_b + l * HDIM, tiss_h, CCLUST);

    // ---- bottom-up cross-level attention ----
    gemm(tiss_h, bu_Wq, bu_bq, tq, CCLUST, HDIM, HDIM, 0, stream);
    gemm(cell_h, bu_Wk, bu_bk, bufA, NCELL, HDIM, HDIM, 0, stream);   // K
    gemm(cell_h, bu_Wv, bu_bv, bufB, NCELL, HDIM, HDIM, 0, stream);   // V
    k_bu_satt<<<g1((long)NCELL * NHEAD), 256, 0, stream>>>(tq, bufA, lab, a_s);
    k_fill<<<g1(CCLUST * NHEAD), 256, 0, stream>>>(m2, CCLUST * NHEAD, -INFINITY);
    hipMemsetAsync(z2, 0, CCLUST * NHEAD * sizeof(float), stream);
    k_seg_max<<<g1((long)NCELL * NHEAD), 256, 0, stream>>>(a_s, lab, m2);
    k_seg_expsum<<<g1((long)NCELL * NHEAD), 256, 0, stream>>>(a_s, lab, m2, z2);
    hipMemsetAsync(tctx, 0, CCLUST * HDIM * sizeof(float), stream);
    k_bu_ctx<<<g1((long)NCELL * NHEAD), 256, 0, stream>>>(a_s, z2, lab, bufB, tctx);
    gemm(tctx, bu_Wo, bu_bo, tiss_h, CCLUST, HDIM, HDIM, 0, stream);

    // ---- top-down (softmax over one key == 1) ----
    gemm(tiss_h, td_Wv, td_bv, tA, CCLUST, HDIM, HDIM, 0, stream);
    gemm(tA, td_Wo, td_bo, ttd, CCLUST, HDIM, HDIM, 0, stream);

    // ---- GRU refinement, exact at cluster level ----
    k_xcat<<<g1(CCLUST * 2 * HDIM), 256, 0, stream>>>(ttd, tiss_h, xcat);
    gemm(xcat, gru_Wih, gru_bih, tgi, CCLUST, 2 * HDIM, 3 * HDIM, 0, stream);
    gemm(ttd, gru_Whh, gru_bhh, tgh, CCLUST, HDIM, 3 * HDIM, 0, stream);
    k_gru<<<g1(CCLUST * HDIM), 256, 0, stream>>>(tgi, tgh, ttd, G);
    k_gather<<<g1((long)NCELL * HDIM), 256, 0, stream>>>(G, lab, cell_h);
  }

  // ---- dual pooling + classifier head ----
  k_pool<<<HDIM, 256, 0, stream>>>(cell_h, NCELL, 1.0f / NCELL, emb, emb + HDIM);
  k_pool<<<HDIM, 256, 0, stream>>>(tiss_h, CCLUST, 1.0f / CCLUST, emb + 2 * HDIM, emb + 3 * HDIM);
  k_mlp<<<1, 256, 0, stream>>>(emb, Wc1, bc1, Wc2, bc2, Wc3, bc3, (float*)d_out);
}